// MaskedLightAdaIN_78477642432611
// MI455X (gfx1250) — compile-verified
//
#include <hip/hip_runtime.h>
#include <hip/hip_bf16.h>
#include <math.h>

#define NB 16
#define NC 64
#define HW 65536
#define HW4 16384       // HW / 4 (float4 count per plane)
#define EPSV 1e-8f

typedef float v2f __attribute__((ext_vector_type(2)));
typedef float v8f __attribute__((ext_vector_type(8)));

// Reduce two independent per-lane stats across a wave32 with one
// V_WMMA_F32_16X16X4_F32. A[M][0..3] = {sa(M), sb(M), sa(M+16), sb(M+16)};
// B[k][j] = 1 iff (k even && j<8) || (k odd && j>=8).
// -> D[M][j<8] = sa(M)+sa(M+16); D[M][j>=8] = sb(M)+sb(M+16).
// Sum the 8 accumulator VGPRs (reduces M within each half-wave), then
// shfl_xor(16) merges halves; column 0 = total(sa), column 8 = total(sb).
__device__ __forceinline__ void wave_reduce_pair(float sa, float sb,
                                                 float& ta, float& tb) {
    const int lane = threadIdx.x & 31;
    v2f A;
    A.x = sa;   // (M=lane&15, K = lane<16 ? 0 : 2)
    A.y = sb;   // (M=lane&15, K = lane<16 ? 1 : 3)
    const bool low = (lane & 15) < 8;
    v2f Bm;
    Bm.x = low ? 1.0f : 0.0f;   // K even rows select columns 0..7
    Bm.y = low ? 0.0f : 1.0f;   // K odd  rows select columns 8..15
    v8f acc = {};
    acc = __builtin_amdgcn_wmma_f32_16x16x4_f32(
        /*neg_a=*/false, A, /*neg_b=*/false, Bm,
        /*c_mod=*/(short)0, acc, /*reuse_a=*/false, /*reuse_b=*/false);
    float r = acc[0] + acc[1] + acc[2] + acc[3] +
              acc[4] + acc[5] + acc[6] + acc[7];
    r += __shfl_xor(r, 16);
    ta = __shfl(r, 0);
    tb = __shfl(r, 8);
}

// ---- Kernel 1: per-batch foreground pixel count ------------------------
__global__ void __launch_bounds__(256)
nfg_kernel(const float* __restrict__ mask, float* __restrict__ nfg) {
    __shared__ float wsum[8];
    const int b = blockIdx.x;
    const int t = threadIdx.x;
    const float4* mp = reinterpret_cast<const float4*>(mask) + (size_t)b * HW4;
    float cnt = 0.0f;
#pragma unroll 4
    for (int i = 0; i < 64; ++i) {
        float4 m = mp[t + i * 256];
        cnt += (m.x >= 0.5f ? 1.0f : 0.0f);
        cnt += (m.y >= 0.5f ? 1.0f : 0.0f);
        cnt += (m.z >= 0.5f ? 1.0f : 0.0f);
        cnt += (m.w >= 0.5f ? 1.0f : 0.0f);
    }
    float ta, tb;
    wave_reduce_pair(cnt, 0.0f, ta, tb);
    if ((t & 31) == 0) wsum[t >> 5] = ta;
    __syncthreads();
    if (t == 0) {
        float s = 0.0f;
#pragma unroll
        for (int w = 0; w < 8; ++w) s += wsum[w];
        nfg[b] = s;
    }
}

// ---- Kernel 2: per-(b,c) raw moments {Σx, Σx², Σx·fg, Σx²·fg} ----------
__global__ void __launch_bounds__(256)
stats_kernel(const float* __restrict__ x, const float* __restrict__ mask,
             float4* __restrict__ stats) {
    __shared__ float wred[8][4];
    const int bc = blockIdx.x;            // b*NC + c
    const int b  = bc >> 6;
    const int t  = threadIdx.x;
    const float4* xp = reinterpret_cast<const float4*>(x) + (size_t)bc * HW4;
    const float4* mp = reinterpret_cast<const float4*>(mask) + (size_t)b * HW4;
    float s = 0.0f, ss = 0.0f, sf = 0.0f, ssf = 0.0f;
#pragma unroll 4
    for (int i = 0; i < 64; ++i) {
        const int idx = t + i * 256;
        float4 v = xp[idx];
        float4 m = mp[idx];
        float f0 = (m.x >= 0.5f) ? 1.0f : 0.0f;
        float f1 = (m.y >= 0.5f) ? 1.0f : 0.0f;
        float f2 = (m.z >= 0.5f) ? 1.0f : 0.0f;
        float f3 = (m.w >= 0.5f) ? 1.0f : 0.0f;
        s += v.x + v.y + v.z + v.w;
        ss  = fmaf(v.x, v.x, ss);  ss  = fmaf(v.y, v.y, ss);
        ss  = fmaf(v.z, v.z, ss);  ss  = fmaf(v.w, v.w, ss);
        sf  = fmaf(f0, v.x, sf);   sf  = fmaf(f1, v.y, sf);
        sf  = fmaf(f2, v.z, sf);   sf  = fmaf(f3, v.w, sf);
        ssf = fmaf(f0 * v.x, v.x, ssf);  ssf = fmaf(f1 * v.y, v.y, ssf);
        ssf = fmaf(f2 * v.z, v.z, ssf);  ssf = fmaf(f3 * v.w, v.w, ssf);
    }
    float ts, tss, tsf, tssf;
    wave_reduce_pair(s,  ss,  ts,  tss);    // totals
    wave_reduce_pair(sf, ssf, tsf, tssf);   // foreground
    if ((t & 31) == 0) {
        const int w = t >> 5;
        wred[w][0] = ts; wred[w][1] = tss; wred[w][2] = tsf; wred[w][3] = tssf;
    }
    __syncthreads();
    if (t == 0) {
        float4 r = make_float4(0.f, 0.f, 0.f, 0.f);
#pragma unroll
        for (int w = 0; w < 8; ++w) {
            r.x += wred[w][0]; r.y += wred[w][1];
            r.z += wred[w][2]; r.w += wred[w][3];
        }
        stats[bc] = r;
    }
}

// ---- Kernel 3: normalize / harmonize -----------------------------------
__global__ void __launch_bounds__(256)
apply_kernel(const float* __restrict__ x, const float* __restrict__ mask,
             const float4* __restrict__ stats, const float* __restrict__ nfg,
             float* __restrict__ out) {
    const int bc = blockIdx.x;
    const int b  = bc >> 6;
    const int t  = threadIdx.x;

    // Uniform per-block finalization (addresses depend only on blockIdx ->
    // scalar loads; a handful of SALU/VALU ops).
    const float4 st = stats[bc];
    const float nf = nfg[b];
    const float nbk = (float)HW - nf;
    const float mu_fg  = st.z / nf;
    const float var_fg = fmaxf((st.w - nf * mu_fg * mu_fg) / (nf - 1.0f), 0.0f);
    const float sig_fg = sqrtf(var_fg);
    const float sum_bg   = st.x - st.z;
    const float sumsq_bg = st.y - st.w;
    const float mu_bg  = sum_bg / nbk;
    const float var_bg = fmaxf((sumsq_bg - nbk * mu_bg * mu_bg) / (nbk - 1.0f), 0.0f);
    const float sig_bg = sqrtf(var_bg);
    const float scale  = sig_fg / (sig_bg + EPSV);

    const float4* xp = reinterpret_cast<const float4*>(x) + (size_t)bc * HW4;
    const float4* mp = reinterpret_cast<const float4*>(mask) + (size_t)b * HW4;
    float4* op = reinterpret_cast<float4*>(out) + (size_t)bc * HW4;
#pragma unroll 4
    for (int i = 0; i < 64; ++i) {
        const int idx = t + i * 256;
        float4 v = xp[idx];
        float4 m = mp[idx];
        float4 o;
        o.x = (m.x >= 0.5f) ? v.x : fmaf(v.x - mu_bg, scale, mu_fg);
        o.y = (m.y >= 0.5f) ? v.y : fmaf(v.y - mu_bg, scale, mu_fg);
        o.z = (m.z >= 0.5f) ? v.z : fmaf(v.z - mu_bg, scale, mu_fg);
        o.w = (m.w >= 0.5f) ? v.w : fmaf(v.w - mu_bg, scale, mu_fg);
        op[idx] = o;
    }
}

extern "C" void kernel_launch(void* const* d_in, const int* in_sizes, int n_in,
                              void* d_out, int out_size, void* d_ws, size_t ws_size,
                              hipStream_t stream) {
    const float* x    = (const float*)d_in[0];   // [16,64,256,256] f32
    const float* mask = (const float*)d_in[1];   // [16,1,256,256] f32
    float* out = (float*)d_out;

    // Workspace: 1024 float4 stats (16 KB) + 16 floats nfg.
    float4* stats = (float4*)d_ws;
    float*  nfg   = (float*)((char*)d_ws + NB * NC * sizeof(float4));

    nfg_kernel<<<NB, 256, 0, stream>>>(mask, nfg);
    stats_kernel<<<NB * NC, 256, 0, stream>>>(x, mask, stats);
    apply_kernel<<<NB * NC, 256, 0, stream>>>(x, mask, stats, nfg, out);
}